// Decoder_9534827397602
// MI455X (gfx1250) — compile-verified
//
#include <hip/hip_runtime.h>
#include <math.h>

#define D     128
#define BATCH 8
#define LSEQ  64
#define BL    (BATCH*LSEQ)   // 512 cells
#define MT    256            // target genes
#define MF    128            // tf genes
#define MR    64             // lig/recp genes

typedef __attribute__((ext_vector_type(2))) float v2f;
typedef __attribute__((ext_vector_type(8))) float v8f;

#define INVSQRTD 0.08838834764831845f   // 1/sqrt(128)

// ---------------------------------------------------------------------------
// Generic fp32 WMMA GEMM: C(M,N) = A(M,K) @ B.  B stored (K,N) row-major, or
// (N,K) row-major when TRANSB=1 (logical B[k,n] = Bm[n*K+k]).
// Each wave computes a 16x32 C tile (two 16x16 accumulators -> two independent
// V_WMMA_F32_16X16X4_F32 chains; A fragment reused for both). K unrolled by 8.
// A-frag per ISA 7.12.2: lane holds M=lane&15; VGPR pair holds K = k0+2*(lane>>4)+{0,1}.
// B-frag mirrored (N=lane&15); C/D: row = tm + 8*(lane>>4) + i, col = tn + (lane&15).
// Requires M%16==0, N%32==0, K%8==0 (true for all call sites here).
// ---------------------------------------------------------------------------
template<int TRANSB>
__global__ void gemm_wmma_f32(const float* __restrict__ A, const float* __restrict__ Bm,
                              float* __restrict__ C, int M, int N, int K)
{
    const int wavesPerBlk = blockDim.x >> 5;
    const int wave = threadIdx.x >> 5;
    const int lane = threadIdx.x & 31;
    const int tilesN = N >> 5;                 // 32-wide column tiles
    const int tiles  = (M >> 4) * tilesN;
    const int t = blockIdx.x * wavesPerBlk + wave;
    if (t >= tiles) return;                    // wave-uniform guard: EXEC stays all-1s
    const int tm   = (t / tilesN) << 4;
    const int tn   = (t % tilesN) << 5;
    const int half = lane >> 4;
    const int l15  = lane & 15;
    const int mr   = tm + l15;                 // A row this lane supplies
    const int nc0  = tn + l15;                 // first B/C column
    const int nc1  = nc0 + 16;                 // second B/C column

    v8f acc0 = {0.f,0.f,0.f,0.f,0.f,0.f,0.f,0.f};
    v8f acc1 = {0.f,0.f,0.f,0.f,0.f,0.f,0.f,0.f};
    const float* Arow = A + (size_t)mr * K;

    for (int k0 = 0; k0 < K; k0 += 8) {
        const int ka0 = k0 + (half << 1);
        const int ka1 = ka0 + 4;
        const v2f a0 = *(const v2f*)(Arow + ka0);
        const v2f a1 = *(const v2f*)(Arow + ka1);
        v2f b00, b01, b10, b11;                // b[kstep][ntile]
        if (TRANSB) {
            const float* Bn0 = Bm + (size_t)nc0 * K;
            const float* Bn1 = Bm + (size_t)nc1 * K;
            b00 = *(const v2f*)(Bn0 + ka0);
            b01 = *(const v2f*)(Bn1 + ka0);
            b10 = *(const v2f*)(Bn0 + ka1);
            b11 = *(const v2f*)(Bn1 + ka1);
        } else {
            const float* Bk0 = Bm + (size_t)ka0 * N;
            const float* Bk1 = Bm + (size_t)ka1 * N;
            b00.x = Bk0[nc0];       b00.y = Bk0[N + nc0];
            b01.x = Bk0[nc1];       b01.y = Bk0[N + nc1];
            b10.x = Bk1[nc0];       b10.y = Bk1[N + nc0];
            b11.x = Bk1[nc1];       b11.y = Bk1[N + nc1];
        }
        acc0 = __builtin_amdgcn_wmma_f32_16x16x4_f32(false, a0, false, b00, (short)0, acc0, false, false);
        acc1 = __builtin_amdgcn_wmma_f32_16x16x4_f32(false, a0, false, b01, (short)0, acc1, false, false);
        acc0 = __builtin_amdgcn_wmma_f32_16x16x4_f32(false, a1, false, b10, (short)0, acc0, false, false);
        acc1 = __builtin_amdgcn_wmma_f32_16x16x4_f32(false, a1, false, b11, (short)0, acc1, false, false);
    }
    const int rbase = tm + (half << 3);
    #pragma unroll
    for (int i = 0; i < 8; ++i) {
        C[(size_t)(rbase + i) * N + nc0] = acc0[i];
        C[(size_t)(rbase + i) * N + nc1] = acc1[i];
    }
}

// ---------------------------------------------------------------------------
// Per-pathway precompute that is not GEMM-shaped.
// ---------------------------------------------------------------------------
struct GeneIn {
    const float *tok, *val, *val2, *disc, *scale;
    const float *wq, *bq, *wk, *bk, *wv, *bv, *wo, *bo;
};
struct PrepWS { float *tokn, *c, *hP, *hN, *bvo; int M; };

__global__ void prep_kernel(GeneIn g, PrepWS w)
{
    __shared__ float gP[D], gN[D], wkbq[D];
    const int tid = threadIdx.x;               // 256 threads
    const int M = w.M;

    // tok_norm: l2-normalize rows of tok[:M]
    for (int m = tid; m < M; m += blockDim.x) {
        float ss = 0.f;
        for (int d = 0; d < D; ++d) { float t = g.tok[m*D + d]; ss += t * t; }
        const float rs = rsqrtf(fmaxf(ss, 1e-12f));
        for (int d = 0; d < D; ++d) w.tokn[m*D + d] = g.tok[m*D + d] * rs;
    }
    if (tid < D) {
        const float vl = g.val[tid];
        gP[tid] = (vl >= 0.f) ? vl : 0.1f * vl;   // LeakyReLU factor when x>=0
        gN[tid] = (vl <= 0.f) ? vl : 0.1f * vl;   // ... when x<0
        float acc = 0.f;                           // wkbq[e] = sum_d wk[e,d]*bq[d]
        for (int d = 0; d < D; ++d) acc += g.wk[tid*D + d] * g.bq[d];
        wkbq[tid] = acc;
    }
    __syncthreads();
    if (tid < D) {
        const float sc = g.scale[0];
        float hp = 0.f, hn = 0.f;
        for (int e = 0; e < D; ++e) {
            const float v2 = g.val2[e*D + tid];
            hp += gP[e] * v2;
            hn += gN[e] * v2;
        }
        w.hP[tid] = hp + sc * gP[tid];
        w.hN[tid] = hn + sc * gN[tid];
        float bv = 0.f;                            // bvo = bv@wo + bo
        for (int e = 0; e < D; ++e) bv += g.bv[e] * g.wo[e*D + tid];
        w.bvo[tid] = bv + g.bo[tid];
    }
    __syncthreads();
    // c_n = tok_norm[n] . wkbq   (== bq . TK_n)
    for (int n = tid; n < M; n += blockDim.x) {
        float acc = 0.f;
        for (int d = 0; d < D; ++d) acc += w.tokn[n*D + d] * wkbq[d];
        w.c[n] = acc;
    }
}

// ---------------------------------------------------------------------------
// v[b,l,m] = softmax_d(x * h±[d]) . disc   (one thread per element, online softmax)
// ---------------------------------------------------------------------------
__global__ void gene_v_kernel(const float* __restrict__ x, const float* __restrict__ hP,
                              const float* __restrict__ hN, const float* __restrict__ disc,
                              float* __restrict__ v, int n)
{
    const int i = blockIdx.x * blockDim.x + threadIdx.x;
    if (i >= n) return;
    const float s = x[i];
    const float* h = (s >= 0.f) ? hP : hN;
    float mx = -3.4e38f, den = 0.f, acc = 0.f;
    for (int d = 0; d < D; ++d) {
        const float z = s * h[d];
        if (z > mx) {
            const float r = expf(mx - z);
            den = den * r + 1.f;
            acc = acc * r + disc[d];
            mx = z;
        } else {
            const float e = expf(z - mx);
            den += e;
            acc += e * disc[d];
        }
    }
    v[i] = acc / den;
}

// ---------------------------------------------------------------------------
// Cell-level causal attention weights: w[b,l,k] = softmax_k(scale*q.k - 1e9*mask)
// grid (BL, 2) pathway in y; 64 threads (one per key).
// ---------------------------------------------------------------------------
__global__ void cellw_kernel(const float* __restrict__ q, const float* __restrict__ k_tf,
                             const float* __restrict__ k_lr, const float* __restrict__ sc_tf,
                             const float* __restrict__ sc_lr, float* __restrict__ w_tf,
                             float* __restrict__ w_lr)
{
    __shared__ float red[LSEQ];
    const int pw = blockIdx.y;
    const float* Km = pw ? k_lr : k_tf;
    const float  sc = pw ? sc_lr[0] : sc_tf[0];
    float* wout = pw ? w_lr : w_tf;
    const int cell = blockIdx.x;
    const int l = cell % LSEQ, b = cell / LSEQ;
    const int k = threadIdx.x;
    const float* qrow = q  + (size_t)cell * D;
    const float* krow = Km + (size_t)(b * LSEQ + k) * D;
    float s = 0.f;
    for (int d = 0; d < D; ++d) s += qrow[d] * krow[d];
    s *= sc;
    if (k > l) s -= 1e9f;                       // matches reference masking exactly
    red[k] = s; __syncthreads();
    for (int off = 32; off > 0; off >>= 1) { if (k < off) red[k] = fmaxf(red[k], red[k+off]); __syncthreads(); }
    const float mx = red[0]; __syncthreads();
    const float e = expf(s - mx);
    red[k] = e; __syncthreads();
    for (int off = 32; off > 0; off >>= 1) { if (k < off) red[k] += red[k+off]; __syncthreads(); }
    wout[(size_t)cell * LSEQ + k] = e / red[0];
}

// ---------------------------------------------------------------------------
// ctx[b,k,:] = TVO^T (r .* v) + X*bvo  where r_n = sum_p x_p A[p,n],
// A = rowsoftmax(invd*(v_p v_n S0[p,n] + v_n c_n)).  128 threads, grid (BL,2).
// ---------------------------------------------------------------------------
struct CtxArgs { const float *v, *x, *S0, *c, *TVO, *bvo; float *ctx; int M; };

__global__ void ctx_kernel(CtxArgs a0, CtxArgs a1)
{
    const CtxArgs a = blockIdx.y ? a1 : a0;
    const int M = a.M;
    const int cell = blockIdx.x;
    const int tid = threadIdx.x;               // 128
    __shared__ float sv[MF], sx[MF], red[MF], rv[MF];
    float vn = 0.f, cn = 0.f, xn = 0.f;
    if (tid < M) {
        vn = a.v[(size_t)cell*M + tid];
        xn = a.x[(size_t)cell*M + tid];
        cn = a.c[tid];
    }
    sv[tid] = (tid < M) ? vn : 0.f;
    sx[tid] = (tid < M) ? xn : 0.f;
    __syncthreads();
    red[tid] = sx[tid]; __syncthreads();
    for (int off = 64; off > 0; off >>= 1) { if (tid < off) red[tid] += red[tid+off]; __syncthreads(); }
    const float X = red[0]; __syncthreads();

    float r = 0.f;
    for (int p = 0; p < M; ++p) {
        const float vp = sv[p];
        const float s = (tid < M) ? INVSQRTD * (vp * vn * a.S0[(size_t)p*M + tid] + vn * cn)
                                  : -3.4e38f;
        red[tid] = s; __syncthreads();
        for (int off = 64; off > 0; off >>= 1) { if (tid < off) red[tid] = fmaxf(red[tid], red[tid+off]); __syncthreads(); }
        const float mx = red[0]; __syncthreads();
        const float e = (tid < M) ? expf(s - mx) : 0.f;
        red[tid] = e; __syncthreads();
        for (int off = 64; off > 0; off >>= 1) { if (tid < off) red[tid] += red[tid+off]; __syncthreads(); }
        r += sx[p] * e / red[0];
        __syncthreads();
    }
    rv[tid] = (tid < M) ? r * vn : 0.f;
    __syncthreads();
    const int d = tid;
    float acc = 0.f;
    for (int n = 0; n < M; ++n) acc += rv[n] * a.TVO[(size_t)n*D + d];
    a.ctx[(size_t)cell*D + d] = acc + X * a.bvo[d];
}

// ---------------------------------------------------------------------------
// W[b,l,:] = sum_k w_tf[b,l,k] ctx_tf[b,k,:] + sum_k w_lr[b,l,k] ctx_lr[b,k,:]
// ---------------------------------------------------------------------------
__global__ void wctx_kernel(const float* __restrict__ w_tf, const float* __restrict__ ctx_tf,
                            const float* __restrict__ w_lr, const float* __restrict__ ctx_lr,
                            float* __restrict__ W)
{
    const int cell = blockIdx.x;
    const int b = cell / LSEQ;
    const int d = threadIdx.x;                 // 128
    float acc = 0.f;
    for (int k = 0; k < LSEQ; ++k)
        acc += w_tf[(size_t)cell*LSEQ + k] * ctx_tf[(size_t)(b*LSEQ + k)*D + d];
    for (int k = 0; k < LSEQ; ++k)
        acc += w_lr[(size_t)cell*LSEQ + k] * ctx_lr[(size_t)(b*LSEQ + k)*D + d];
    W[(size_t)cell*D + d] = acc;
}

// ---------------------------------------------------------------------------
// out[b,l,m] = sum_j A_tgt[m,j] * v_j * (TVO_tgt[j].W[b,l]) + bvo_tgt.W[b,l]
// 256 threads (one per output gene m), online softmax over j.
// ---------------------------------------------------------------------------
__global__ void out_kernel(const float* __restrict__ W, const float* __restrict__ v_t,
                           const float* __restrict__ S0, const float* __restrict__ cvec,
                           const float* __restrict__ TVO, const float* __restrict__ bvo,
                           float* __restrict__ out)
{
    __shared__ float Wl[D], tl[MT], svt[MT], scv[MT], red[MT];
    const int cell = blockIdx.x;
    const int tid = threadIdx.x;               // 256
    if (tid < D) Wl[tid] = W[(size_t)cell*D + tid];
    svt[tid] = v_t[(size_t)cell*MT + tid];
    scv[tid] = cvec[tid];
    __syncthreads();
    float u = 0.f;
    for (int d = 0; d < D; ++d) u += TVO[(size_t)tid*D + d] * Wl[d];
    tl[tid] = svt[tid] * u;
    red[tid] = (tid < D) ? bvo[tid] * Wl[tid] : 0.f;
    __syncthreads();
    for (int off = 128; off > 0; off >>= 1) { if (tid < off) red[tid] += red[tid+off]; __syncthreads(); }
    const float bvoW = red[0];
    __syncthreads();
    const float vm = svt[tid];
    float mx = -3.4e38f, den = 0.f, acc = 0.f;
    for (int j = 0; j < MT; ++j) {
        const float vj = svt[j];
        const float s = INVSQRTD * (vm * vj * S0[(size_t)tid*MT + j] + vj * scv[j]);
        if (s > mx) {
            const float rr = expf(mx - s);
            den = den * rr + 1.f;
            acc = acc * rr + tl[j];
            mx = s;
        } else {
            const float e = expf(s - mx);
            den += e;
            acc += e * tl[j];
        }
    }
    out[(size_t)cell*MT + tid] = acc / den + bvoW;
}

// ---------------------------------------------------------------------------
// Host-side launch. Inputs follow setup_inputs() dict insertion order, with the
// three nested _gene_params dicts flattened in their insertion order (13 each).
// ---------------------------------------------------------------------------
extern "C" void kernel_launch(void* const* d_in, const int* in_sizes, int n_in,
                              void* d_out, int out_size, void* d_ws, size_t ws_size,
                              hipStream_t stream)
{
    (void)in_sizes; (void)n_in; (void)out_size; (void)ws_size;
    const float* target_exp = (const float*)d_in[0];
    const float* tf_exp     = (const float*)d_in[1];
    const float* lr_exp     = (const float*)d_in[2];
    const float* ce_tgt     = (const float*)d_in[3];
    const float* ce_tf      = (const float*)d_in[4];
    const float* ce_lr      = (const float*)d_in[5];
    auto gene = [&](int base) {
        GeneIn g;
        g.tok   = (const float*)d_in[base + 0];
        g.val   = (const float*)d_in[base + 1];
        g.val2  = (const float*)d_in[base + 2];
        g.disc  = (const float*)d_in[base + 3];
        g.scale = (const float*)d_in[base + 4];
        g.wq = (const float*)d_in[base + 5];  g.bq = (const float*)d_in[base + 6];
        g.wk = (const float*)d_in[base + 7];  g.bk = (const float*)d_in[base + 8];
        g.wv = (const float*)d_in[base + 9];  g.bv = (const float*)d_in[base + 10];
        g.wo = (const float*)d_in[base + 11]; g.bo = (const float*)d_in[base + 12];
        return g;
    };
    const GeneIn g_t = gene(6), g_f = gene(19), g_r = gene(32);
    const float* sc_tf = (const float*)d_in[45];
    const float* sc_lr = (const float*)d_in[46];

    // workspace carve-up (~4.3 MB of fp32)
    float* ws = (float*)d_ws;
    size_t o = 0;
    auto take = [&](size_t n) { float* p = ws + o; o += n; return p; };
    struct PD { float *tokn, *TQ, *TK, *TV, *TVO, *S0, *c, *hP, *hN, *bvo, *v; int M; };
    auto mkpd = [&](int M) {
        PD p; p.M = M;
        p.tokn = take((size_t)M*D); p.TQ = take((size_t)M*D); p.TK = take((size_t)M*D);
        p.TV = take((size_t)M*D);   p.TVO = take((size_t)M*D); p.S0 = take((size_t)M*M);
        p.c  = take(M); p.hP = take(D); p.hN = take(D); p.bvo = take(D);
        p.v  = take((size_t)BL*M);
        return p;
    };
    PD p_t = mkpd(MT), p_f = mkpd(MF), p_r = mkpd(MR);
    float* qbuf  = take((size_t)BL*D);
    float* ktf   = take((size_t)BL*D);
    float* klr   = take((size_t)BL*D);
    float* wtf   = take((size_t)BL*LSEQ);
    float* wlr   = take((size_t)BL*LSEQ);
    float* ctxtf = take((size_t)BL*D);
    float* ctxlr = take((size_t)BL*D);
    float* Wbuf  = take((size_t)BL*D);

    // 1) per-pathway scalar precompute (tok_norm, h±, c, bvo)
    auto prep = [&](const GeneIn& g, PD& p) {
        PrepWS w; w.tokn = p.tokn; w.c = p.c; w.hP = p.hP; w.hN = p.hN; w.bvo = p.bvo; w.M = p.M;
        prep_kernel<<<1, 256, 0, stream>>>(g, w);
    };
    prep(g_t, p_t); prep(g_f, p_f); prep(g_r, p_r);

    // 2) WMMA GEMMs: pathway derived matrices + cell projections
    auto gemm = [&](const float* A, const float* Bm, float* C, int M, int N, int K, int transB) {
        const int tiles = (M / 16) * (N / 32);
        const int blocks = (tiles + 3) / 4;
        if (transB)
            gemm_wmma_f32<1><<<blocks, 128, 0, stream>>>(A, Bm, C, M, N, K);
        else
            gemm_wmma_f32<0><<<blocks, 128, 0, stream>>>(A, Bm, C, M, N, K);
    };
    auto derive = [&](const GeneIn& g, PD& p) {
        gemm(p.tokn, g.wq, p.TQ, p.M, D, D, 0);
        gemm(p.tokn, g.wk, p.TK, p.M, D, D, 0);
        gemm(p.tokn, g.wv, p.TV, p.M, D, D, 0);
        gemm(p.TV,   g.wo, p.TVO, p.M, D, D, 0);
        gemm(p.TQ,   p.TK, p.S0, p.M, p.M, D, 1);   // S0 = TQ @ TK^T
    };
    derive(g_t, p_t); derive(g_f, p_f); derive(g_r, p_r);
    gemm(target_exp, ce_tgt, qbuf, BL, D, MT, 0);
    gemm(tf_exp,     ce_tf,  ktf,  BL, D, MF, 0);
    gemm(lr_exp,     ce_lr,  klr,  BL, D, MR, 0);

    // 3) per-element gene value scalars
    auto vlaunch = [&](const float* x, const GeneIn& g, PD& p) {
        const int n = BL * p.M;
        gene_v_kernel<<<(n + 255) / 256, 256, 0, stream>>>(x, p.hP, p.hN, g.disc, p.v, n);
    };
    vlaunch(target_exp, g_t, p_t); vlaunch(tf_exp, g_f, p_f); vlaunch(lr_exp, g_r, p_r);

    // 4) causal cell-attention weights (both pathways)
    cellw_kernel<<<dim3(BL, 2), LSEQ, 0, stream>>>(qbuf, ktf, klr, sc_tf, sc_lr, wtf, wlr);

    // 5) per-cell ctx for tf and lr pathways
    CtxArgs a_tf { p_f.v, tf_exp, p_f.S0, p_f.c, p_f.TVO, p_f.bvo, ctxtf, MF };
    CtxArgs a_lr { p_r.v, lr_exp, p_r.S0, p_r.c, p_r.TVO, p_r.bvo, ctxlr, MR };
    ctx_kernel<<<dim3(BL, 2), 128, 0, stream>>>(a_tf, a_lr);

    // 6) weighted context combine
    wctx_kernel<<<BL, 128, 0, stream>>>(wtf, ctxtf, wlr, ctxlr, Wbuf);

    // 7) final output via target-gene attention applied to projected context
    out_kernel<<<BL, 256, 0, stream>>>(Wbuf, p_t.v, p_t.S0, p_t.c, p_t.TVO, p_t.bvo,
                                       (float*)d_out);
}